// Network_20134806684319
// MI455X (gfx1250) — compile-verified
//
#include <hip/hip_runtime.h>
#include <math.h>

typedef __attribute__((ext_vector_type(2))) float v2f;
typedef __attribute__((ext_vector_type(8))) float v8f;

__device__ __forceinline__ v8f wmma4(v2f a, v2f b, v8f c) {
  // D = A(16x4 f32) * B(4x16 f32) + C(16x16 f32)
  return __builtin_amdgcn_wmma_f32_16x16x4_f32(
      /*neg_a=*/false, a, /*neg_b=*/false, b,
      /*c_mod=*/(short)0, c, /*reuse_a=*/false, /*reuse_b=*/false);
}

// One block = 16 output rows of one K-slice (blockIdx.y). 8 waves split the
// slice; each wave runs WMMA f32 16x16x4 with B = activation chunk broadcast
// across all 16 columns, so every column of D carries the same dot-product
// partial. Loop bounds are wave-uniform -> readfirstlane pushes loop control
// to SALU (no EXEC churn around WMMA). #pragma unroll 4 lets the scheduler
// keep up to 32 b64 loads in flight; global_prefetch_b8 runs ~1KB ahead.
__global__ __launch_bounds__(256) void gemv16_wmma(const float* __restrict__ W,
                                                   const float* __restrict__ act,
                                                   float* __restrict__ out,
                                                   int K) {
  __shared__ float red[8 * 16];
  const int lane = threadIdx.x & 31;
  const int wave = threadIdx.x >> 5;
  const int r    = lane & 15;      // A-matrix M index within tile
  const int h    = lane >> 4;      // K half-select: lanes 16-31 take K+2..3
  const int row0 = blockIdx.x << 4;
  const int M    = gridDim.x << 4;

  const int nchunks = (int)gridDim.y * 8;        // total wave-chunks over K
  const int chunk   = (int)blockIdx.y * 8 + wave;
  const int Kc = K / nchunks;                     // multiple of 64 in all cases
  // Wave-uniform bounds -> SGPRs (scalar loop control, EXEC stays all-1s).
  const int kb = __builtin_amdgcn_readfirstlane(chunk * Kc);
  const int ke = __builtin_amdgcn_readfirstlane(kb + Kc);

  const float* wp = W + (size_t)(row0 + r) * (size_t)K + (h << 1);
  const float* ap = act + (h << 1);

  v8f c = {};
#pragma unroll 4
  for (int k = kb; k < ke; k += 16) {
    __builtin_prefetch(wp + k + 256, 0, 3);      // ~1KB ahead
    v2f w0 = *(const v2f*)(wp + k);
    v2f w1 = *(const v2f*)(wp + k + 4);
    v2f w2 = *(const v2f*)(wp + k + 8);
    v2f w3 = *(const v2f*)(wp + k + 12);
    v2f a0 = *(const v2f*)(ap + k);
    v2f a1 = *(const v2f*)(ap + k + 4);
    v2f a2 = *(const v2f*)(ap + k + 8);
    v2f a3 = *(const v2f*)(ap + k + 12);
    c = wmma4(w0, a0, c);
    c = wmma4(w1, a1, c);
    c = wmma4(w2, a2, c);
    c = wmma4(w3, a3, c);
  }

  // C/D layout: lane 0 holds rows row0+0..7 in c[0..7]; lane 16 holds rows +8..15.
  if (lane == 0) {
#pragma unroll
    for (int i = 0; i < 8; ++i) red[wave * 16 + i] = c[i];
  } else if (lane == 16) {
#pragma unroll
    for (int i = 0; i < 8; ++i) red[wave * 16 + 8 + i] = c[i];
  }
  __syncthreads();

  if (threadIdx.x < 16) {
    float s = 0.f;
#pragma unroll
    for (int w = 0; w < 8; ++w) s += red[w * 16 + threadIdx.x];
    out[(size_t)blockIdx.y * M + row0 + threadIdx.x] = s;
  }
}

// ws regions (ns = K-split count):
//   DG: ns*8192 | CA3A, CA3B, CA3C, CA1, EC: ns*4096 each (in that order)
// out layout: [0,4096) EC | [4096,12288) DG | [12288,16384) CA3 | [16384,20480) CA1
__global__ __launch_bounds__(256) void finalize_kernel(const float* __restrict__ ws,
                                                       const float* __restrict__ x_EC,
                                                       float* __restrict__ out,
                                                       int ns) {
  const int i = blockIdx.x * 256 + threadIdx.x;  // 0..20479
  const int bDG   = 0;
  const int bCA3A = ns * 8192;
  const int bCA3B = bCA3A + ns * 4096;
  const int bCA3C = bCA3B + ns * 4096;
  const int bCA1  = bCA3C + ns * 4096;
  const int bEC   = bCA1 + ns * 4096;

  float v = 0.f;
  if (i < 4096) {
    for (int s = 0; s < ns; ++s) v += ws[bEC + s * 4096 + i];
    v += x_EC[i];
  } else if (i < 12288) {
    const int j = i - 4096;
    for (int s = 0; s < ns; ++s) v += ws[bDG + s * 8192 + j];
  } else if (i < 16384) {
    const int j = i - 12288;
    for (int s = 0; s < ns; ++s) v += ws[bCA3A + s * 4096 + j];
    for (int s = 0; s < ns; ++s) v += ws[bCA3B + s * 4096 + j];
    for (int s = 0; s < ns; ++s) v += ws[bCA3C + s * 4096 + j];
  } else {
    const int j = i - 16384;
    for (int s = 0; s < ns; ++s) v += ws[bCA1 + s * 4096 + j];
  }
  out[i] = 1.0f / (1.0f + expf(-v));
}

extern "C" void kernel_launch(void* const* d_in, const int* in_sizes, int n_in,
                              void* d_out, int out_size, void* d_ws, size_t ws_size,
                              hipStream_t stream) {
  const float* act_EC    = (const float*)d_in[0];
  const float* act_DG    = (const float*)d_in[1];
  const float* act_CA3   = (const float*)d_in[2];
  const float* act_CA1   = (const float*)d_in[3];
  const float* w_EC_DG   = (const float*)d_in[4];   // [8192, 4096]
  const float* w_DG_CA3  = (const float*)d_in[5];   // [4096, 8192]
  const float* w_CA3_CA3 = (const float*)d_in[6];   // [4096, 4096]
  const float* w_EC_CA3  = (const float*)d_in[7];   // [4096, 4096]
  const float* w_CA3_CA1 = (const float*)d_in[8];   // [4096, 4096]
  const float* w_CA1_EC  = (const float*)d_in[9];   // [4096, 4096]
  const float* x_EC      = (const float*)d_in[10];
  float* out = (float*)d_out;
  float* ws  = (float*)d_ws;

  // Pick the K-split (more blocks in flight) that fits the workspace.
  // Bytes needed: ns * (8192 + 5*4096) * 4 = ns * 114688.
  int ns = 4;
  if (ws_size < (size_t)4 * 114688) ns = 2;
  if (ws_size < (size_t)2 * 114688) ns = 1;

  float* wsDG   = ws;
  float* wsCA3A = wsDG   + ns * 8192;
  float* wsCA3B = wsCA3A + ns * 4096;
  float* wsCA3C = wsCA3B + ns * 4096;
  float* wsCA1  = wsCA3C + ns * 4096;
  float* wsEC   = wsCA1  + ns * 4096;

  const dim3 blk(256);
  gemv16_wmma<<<dim3(8192 / 16, ns), blk, 0, stream>>>(w_EC_DG,   act_EC,  wsDG,   4096);
  gemv16_wmma<<<dim3(4096 / 16, ns), blk, 0, stream>>>(w_DG_CA3,  act_DG,  wsCA3A, 8192);
  gemv16_wmma<<<dim3(4096 / 16, ns), blk, 0, stream>>>(w_CA3_CA3, act_CA3, wsCA3B, 4096);
  gemv16_wmma<<<dim3(4096 / 16, ns), blk, 0, stream>>>(w_EC_CA3,  act_EC,  wsCA3C, 4096);
  gemv16_wmma<<<dim3(4096 / 16, ns), blk, 0, stream>>>(w_CA3_CA1, act_CA3, wsCA1,  4096);
  gemv16_wmma<<<dim3(4096 / 16, ns), blk, 0, stream>>>(w_CA1_EC,  act_CA1, wsEC,   4096);
  finalize_kernel<<<dim3(20480 / 256), blk, 0, stream>>>(ws, x_EC, out, ns);
}